// VarianceLoss_48601849921610
// MI455X (gfx1250) — compile-verified
//
#include <hip/hip_runtime.h>

typedef __attribute__((ext_vector_type(2))) float v2f;
typedef __attribute__((ext_vector_type(8))) float v8f;

#define BB 8
#define NN 64
#define KK 32
#define HH 256
#define WW 256
#define HW (HH * WW)

// Async global->LDS copy of 16 bytes per enabled lane (CDNA5 ASYNCcnt path).
__device__ __forceinline__ void async_copy_b128(unsigned lds_byte_addr,
                                                const void* gptr) {
    unsigned long long ga = (unsigned long long)gptr;
    asm volatile("global_load_async_to_lds_b128 %0, %1, off"
                 :
                 : "v"(lds_byte_addr), "v"(ga)
                 : "memory");
}

__device__ __forceinline__ void wait_async_zero() {
    asm volatile("s_wait_asynccnt 0x0" ::: "memory");
}

// ---------------------------------------------------------------------------
// Zero the scalar output (harness poisons once; we must init every call).
// ---------------------------------------------------------------------------
__global__ void vl_zero_kernel(float* out) {
    if (threadIdx.x == 0 && blockIdx.x == 0) out[0] = 0.0f;
}

// ---------------------------------------------------------------------------
// Separable gaussian axes: Ex[b,n,w] = exp(-0.5*((xw-cx)/sx)^2),
//                          Ey[b,n,h] = exp(-0.5*((yh-cy)/sy)^2)
// One block per (b,n); 256 threads cover both the 256 w's and 256 h's.
// ---------------------------------------------------------------------------
__global__ __launch_bounds__(256)
void vl_gauss_axes_kernel(const float* __restrict__ loc,
                          float* __restrict__ Ex, float* __restrict__ Ey) {
    const int bn = blockIdx.x;          // 0 .. B*N-1
    const int t  = threadIdx.x;         // 0 .. 255 (== h and w index)
    const float cx = loc[bn * 4 + 0];
    const float cy = loc[bn * 4 + 1];
    const float sx = loc[bn * 4 + 2];
    const float sy = loc[bn * 4 + 3];
    const float p  = ((float)t + 0.5f) * (1.0f / 256.0f);
    const float dx = (p - cx) / sx;
    const float dy = (p - cy) / sy;
    Ex[bn * WW + t] = __expf(-0.5f * dx * dx);
    Ey[bn * HH + t] = __expf(-0.5f * dy * dy);
}

// ---------------------------------------------------------------------------
// Main kernel: per batch, per 128-hw tile:
//   S = proto_tile[16x32] @ masks^T[32x64] via V_WMMA_F32_16X16X4_F32
//   m = sigmoid(S) * Ey[n,h] * Ex[n,w] * conf[n]
//   final_conf[hw] = 1 - (sum_n m^2) / (sum_n m + 1e-5)
// 256 threads = 8 wave32; each wave owns a 16-row (hw) WMMA tile.
// Block covers 128 consecutive hw => h is constant for the whole block.
// All LDS staging uses GLOBAL_LOAD_ASYNC_TO_LDS_B128 (no VGPR round-trip).
// ---------------------------------------------------------------------------
__global__ __launch_bounds__(256)
void vl_mask_conf_kernel(const float* __restrict__ proto,
                         const float* __restrict__ masks,
                         const float* __restrict__ conf,
                         const float* __restrict__ Ex,
                         const float* __restrict__ Ey,
                         float* __restrict__ finalConf) {
    __shared__ float sMask[NN * KK];     //  8 KB : masks[b] (64x32)
    __shared__ float sEx[NN * 128];      // 32 KB : Ex slice for this block's w range
    __shared__ float sA[8 * 16 * KK];    // 16 KB : per-wave 16x32 proto tiles

    const int tid  = threadIdx.x;
    const int wave = tid >> 5;
    const int lane = tid & 31;
    const int half = lane >> 4;          // 0: lanes 0-15, 1: lanes 16-31
    const int l15  = lane & 15;

    const int blk    = blockIdx.x & 511;   // hw-tile index within batch
    const int b      = blockIdx.x >> 9;
    const int hwbase = blk * 128;          // multiple of 128 => h constant
    const int h      = hwbase >> 8;
    const int wbase  = hwbase & 255;       // 0 or 128

    // LDS byte addresses (low 32 bits of the generic pointer == LDS offset).
    const unsigned ldsMask = (unsigned)(unsigned long long)(void*)sMask;
    const unsigned ldsEx   = (unsigned)(unsigned long long)(void*)sEx;
    const unsigned ldsA    = (unsigned)(unsigned long long)(void*)sA;

    // --- Stage masks[b] (8 KB, contiguous): 512 float4s, 2 per thread. ---
    {
        const float4* m4 = (const float4*)(masks + (size_t)b * NN * KK);
#pragma unroll
        for (int r = 0; r < 2; ++r) {
            const int i = r * 256 + tid;
            async_copy_b128(ldsMask + i * 16, m4 + i);
        }
    }

    // --- Stage Ex[b, n=0..63, w=wbase..+32): 64 rows x 8 float4s. ---
    {
#pragma unroll
        for (int r = 0; r < 2; ++r) {
            const int i   = r * 256 + tid;      // 0..511
            const int n   = i >> 3;
            const int seg = i & 7;
            const float4* g4 =
                (const float4*)(Ex + ((size_t)(b * NN + n)) * WW + wbase) + seg;
            async_copy_b128(ldsEx + i * 16, g4);
        }
    }

    // --- Stage per-wave 16x32 proto tile (2 KB contiguous in proto). ---
    const int rowbase = hwbase + wave * 16;
    {
        const float4* p4 = (const float4*)(proto + ((size_t)b * HW + rowbase) * KK);
#pragma unroll
        for (int i = 0; i < 4; ++i) {
            const int idx = i * 32 + lane;      // float4 index within tile
            async_copy_b128(ldsA + wave * 2048 + idx * 16, p4 + idx);
        }
    }

    wait_async_zero();       // drain ASYNCcnt before cross-wave consumption
    __syncthreads();

    // GEMM: D[16x16] += A[16x4] x B[4x16], K=32 -> 8 steps, 4 N-tiles.
    // f32 A layout : lane -> M = lane&15 ; VGPR v -> K = (lane>>4)*2 + v
    // f32 B layout : lane -> N = lane&15 ; VGPR v -> K = (lane>>4)*2 + v
    v8f acc[4];
    const v8f zero8 = {0.f, 0.f, 0.f, 0.f, 0.f, 0.f, 0.f, 0.f};
    acc[0] = zero8; acc[1] = zero8; acc[2] = zero8; acc[3] = zero8;

    const float* A = sA + wave * 16 * KK;
#pragma unroll
    for (int s = 0; s < 8; ++s) {
        const int kk = s * 4 + half * 2;
        v2f a;
        a.x = A[l15 * KK + kk];
        a.y = A[l15 * KK + kk + 1];
#pragma unroll
        for (int nt = 0; nt < 4; ++nt) {
            v2f bf;
            bf.x = sMask[(nt * 16 + l15) * KK + kk];
            bf.y = sMask[(nt * 16 + l15) * KK + kk + 1];
            acc[nt] = __builtin_amdgcn_wmma_f32_16x16x4_f32(
                false, a, false, bf, (short)0, acc[nt], false, false);
        }
    }

    // Epilogue. D layout: lane -> N = lane&15 (col), VGPR v -> M = v + 8*half.
    // Per lane/nt: n = nt*16 + l15 (fixed over v) -> preload Ey*conf.
    float eycf[4];
#pragma unroll
    for (int nt = 0; nt < 4; ++nt) {
        const int n = nt * 16 + l15;
        eycf[nt] = Ey[(b * NN + n) * HH + h] * conf[b * NN + n];
    }

#pragma unroll
    for (int v = 0; v < 8; ++v) {
        const int hw = rowbase + v + half * 8;   // hw row this element belongs to
        const int wo = hw & 127;                 // offset in block's w-slice
        float sm = 0.f, sm2 = 0.f;
#pragma unroll
        for (int nt = 0; nt < 4; ++nt) {
            const int n = nt * 16 + l15;
            const float x   = acc[nt][v];
            const float sig = 1.0f / (1.0f + __expf(-x));
            const float m   = sig * eycf[nt] * sEx[n * 128 + wo];
            sm  += m;
            sm2 += m * m;
        }
        // Reduce over the 16 lanes of each half (they hold n-partials for the
        // same hw row): butterfly with offsets 1,2,4,8 stays within halves.
#pragma unroll
        for (int off = 1; off < 16; off <<= 1) {
            sm  += __shfl_xor(sm, off, 32);
            sm2 += __shfl_xor(sm2, off, 32);
        }
        if (l15 == 0)
            finalConf[(size_t)b * HW + hw] = 1.0f - sm2 / (sm + 1e-5f);
    }
}

// ---------------------------------------------------------------------------
// Final reduction: per pixel (h,w):
//   tot = sum_b fc[b]; wm[c] = sum_b orig[b,c]*fc[b]
//   loss += sum_{b,c} (orig[b,c]-wm[c])^2 * fc[b] / tot
// Block tree-reduce then one atomicAdd into the scalar output.
// ---------------------------------------------------------------------------
__global__ __launch_bounds__(256)
void vl_reduce_kernel(const float* __restrict__ original,
                      const float* __restrict__ finalConf,
                      float* __restrict__ out) {
    __shared__ float red[256];
    const int idx = blockIdx.x * 256 + threadIdx.x;   // pixel index

    float cb[BB];
    float tot = 0.f;
#pragma unroll
    for (int b = 0; b < BB; ++b) {
        cb[b] = finalConf[(size_t)b * HW + idx];
        tot += cb[b];
    }

    float loss = 0.f;
#pragma unroll
    for (int c = 0; c < 3; ++c) {
        float o[BB];
        float wm = 0.f;
#pragma unroll
        for (int b = 0; b < BB; ++b) {
            o[b] = original[((size_t)b * 3 + c) * HW + idx];
            wm += o[b] * cb[b];
        }
#pragma unroll
        for (int b = 0; b < BB; ++b) {
            const float d = o[b] - wm;
            loss += d * d * cb[b];
        }
    }
    loss /= tot;

    red[threadIdx.x] = loss;
    __syncthreads();
    for (int s = 128; s > 0; s >>= 1) {
        if (threadIdx.x < s) red[threadIdx.x] += red[threadIdx.x + s];
        __syncthreads();
    }
    if (threadIdx.x == 0) atomicAdd(out, red[0]);
}

// ---------------------------------------------------------------------------
// Launch
// ---------------------------------------------------------------------------
extern "C" void kernel_launch(void* const* d_in, const int* in_sizes, int n_in,
                              void* d_out, int out_size, void* d_ws, size_t ws_size,
                              hipStream_t stream) {
    const float* original = (const float*)d_in[0];  // [B,3,H,W]
    const float* loc      = (const float*)d_in[1];  // [B,N,4]
    const float* masks    = (const float*)d_in[2];  // [B,N,K]
    const float* proto    = (const float*)d_in[3];  // [B,H,W,K]
    const float* conf     = (const float*)d_in[4];  // [B,N]
    float* out = (float*)d_out;                     // scalar

    // Workspace layout (floats): Ex [B*N*W] | Ey [B*N*H] | finalConf [B*H*W]
    float* Ex = (float*)d_ws;
    float* Ey = Ex + (size_t)BB * NN * WW;
    float* fc = Ey + (size_t)BB * NN * HH;

    vl_zero_kernel<<<1, 32, 0, stream>>>(out);
    vl_gauss_axes_kernel<<<BB * NN, 256, 0, stream>>>(loc, Ex, Ey);
    vl_mask_conf_kernel<<<BB * (HW / 128), 256, 0, stream>>>(proto, masks, conf,
                                                             Ex, Ey, fc);
    vl_reduce_kernel<<<HW / 256, 256, 0, stream>>>(original, fc, out);
}